// BigramLMLinear_41532333752648
// MI455X (gfx1250) — compile-verified
//
#include <hip/hip_runtime.h>

// Z[b,t,:] = W[:, idx[b,t]]  — column gather from W (16384x16384 fp32).
// Inverted into a scatter: bucket tokens by 32-column window; each block
// TDM-DMAs a 256x32 tile of W into LDS (hardware-padded to stride 33 for
// conflict-free column reads), then streams 1KB coalesced NT stores per
// matching token. ~2 GiB total HBM traffic => ~86us roofline @ 23.3 TB/s.

#define VOCAB   16384
#define WIN     32              // columns per window
#define NWIN    (VOCAB / WIN)   // 512
#define VT      256             // rows per tile
#define TSTRIDE 33              // LDS row stride in floats (TDM pad: 32+1)

typedef unsigned u32x4 __attribute__((ext_vector_type(4)));
typedef int      i32x4 __attribute__((ext_vector_type(4)));
typedef int      i32x8 __attribute__((ext_vector_type(8)));

// ---------------------------------------------------------------------------
// Main kernel (emitted first so the disasm snippet shows the TDM sequence).
// Block = (column window w, row chunk v0). TDM-load the W tile into LDS with
// hardware padding, then for each token in the window stream out[t][v0+tid]
// as coalesced NT stores.
// ---------------------------------------------------------------------------
__global__ __launch_bounds__(256)
void gather_scatter_kernel(const float* __restrict__ W,
                           const unsigned* __restrict__ offsets,
                           const unsigned* __restrict__ entries,
                           float* __restrict__ out)
{
    __shared__ float tile[VT * TSTRIDE];   // 33.8 KB, rows padded by TDM

    const int w   = blockIdx.x;            // 0..NWIN-1
    const int v0  = blockIdx.y * VT;       // 0..VOCAB-VT
    const int c0  = w * WIN;
    const int tid = threadIdx.x;           // 0..255

    if (tid < 32) {  // wave 0 issues the tensor DMA; EXEC ignored by TDM
        const unsigned long long ga =
            (unsigned long long)(const void*)(W + (size_t)v0 * VOCAB + c0);
        const unsigned lds = (unsigned)(unsigned long long)(void*)&tile[0];

        // D# group 0: count=1 | lds_addr | global_addr[56:0] | type=2
        u32x4 g0;
        g0.x = 1u;                                     // count=1, user desc
        g0.y = lds;                                    // LDS byte address
        g0.z = (unsigned)ga;                           // global_addr[31:0]
        g0.w = (unsigned)((ga >> 32) & 0x01FFFFFFu)    // global_addr[56:32]
             | (2u << 30);                             // type = 2 ("image")

        // D# group 1
        i32x8 g1;
        g1[0] = (int)((2u << 16)      // data_size = 2 -> 4 bytes
                    | (1u << 20)      // pad_enable
                    | (4u << 22));    // pad_interval = 4 -> every 32 DWORDs
                                      // pad_amount  = 0 -> 1 DWORD pad
        g1[1] = (int)(((unsigned)VOCAB & 0xFFFFu) << 16);   // tensor_dim0 lo16
        g1[2] = (int)(((unsigned)VOCAB >> 16)               // tensor_dim0 hi16
                    | (((unsigned)VOCAB & 0xFFFFu) << 16)); // tensor_dim1 lo16
        g1[3] = (int)(((unsigned)VOCAB >> 16)               // tensor_dim1 hi16
                    | ((unsigned)WIN << 16));               // tile_dim0 = 32
        g1[4] = (int)VT;              // tile_dim1 = 256; tile_dim2 = 0
        g1[5] = (int)VOCAB;           // tensor_dim0_stride[31:0] = 16384
        g1[6] = 0;                    // stride hi bits, dim1_stride lo
        g1[7] = 0;

        i32x4 g2 = {0, 0, 0, 0};      // 2D tile: higher dims unused
        i32x4 g3 = {0, 0, 0, 0};
        i32x8 g4 = {0, 0, 0, 0, 0, 0, 0, 0};  // clang-23 extra arg: inert

        __builtin_amdgcn_tensor_load_to_lds(g0, g1, g2, g3, g4, 0 /*cpol*/);
        __builtin_amdgcn_s_wait_tensorcnt(0);
    }
    __syncthreads();

    const unsigned beg = offsets[w];
    const unsigned end = offsets[w + 1];
    for (unsigned e = beg; e < end; ++e) {
        const unsigned p  = entries[e];
        const unsigned t  = p >> 14;          // token id
        const unsigned cc = p & (WIN - 1);    // column within window
        // stride-33 LDS column read: conflict-free across a wave
        const float val = tile[tid * TSTRIDE + cc];
        __builtin_nontemporal_store(val, out + (size_t)t * VOCAB + v0 + tid);
    }
}

// ---------------------------------------------------------------------------
// Kernel 1: one block. Histogram tokens into 512 column-windows, prefix-scan,
// then scatter packed (token<<14 | column) entries into compact buckets.
// ---------------------------------------------------------------------------
__global__ __launch_bounds__(512)
void build_buckets_kernel(const int* __restrict__ idx, int ntok,
                          unsigned* __restrict__ offsets,   // [NWIN+1]
                          unsigned* __restrict__ entries)   // [ntok]
{
    __shared__ unsigned cnt[NWIN];
    __shared__ unsigned scan[NWIN];
    const int tid = threadIdx.x;           // blockDim.x == NWIN == 512

    cnt[tid] = 0;
    __syncthreads();

    for (int t = tid; t < ntok; t += 512) {
        int c = idx[t];
        atomicAdd(&cnt[c >> 5], 1u);
    }
    __syncthreads();

    // Hillis-Steele inclusive scan over 512 counts
    scan[tid] = cnt[tid];
    __syncthreads();
    for (int off = 1; off < NWIN; off <<= 1) {
        unsigned v = 0;
        if (tid >= off) v = scan[tid - off];
        __syncthreads();
        if (tid >= off) scan[tid] += v;
        __syncthreads();
    }

    offsets[tid + 1] = scan[tid];
    if (tid == 0) offsets[0] = 0;

    // cursors = exclusive scan (reuse cnt[])
    unsigned excl = scan[tid] - cnt[tid];
    __syncthreads();
    cnt[tid] = excl;
    __syncthreads();

    for (int t = tid; t < ntok; t += 512) {
        int c = idx[t];
        unsigned pos = atomicAdd(&cnt[c >> 5], 1u);
        entries[pos] = (unsigned(t) << 14) | unsigned(c);
    }
}

// ---------------------------------------------------------------------------
extern "C" void kernel_launch(void* const* d_in, const int* in_sizes, int n_in,
                              void* d_out, int out_size, void* d_ws, size_t ws_size,
                              hipStream_t stream) {
    const int*   idx = (const int*)d_in[0];    // [8*2048] token indices
    const float* W   = (const float*)d_in[1];  // [VOCAB*VOCAB]
    float*       out = (float*)d_out;          // [8*2048*VOCAB]
    const int ntok   = in_sizes[0];

    unsigned* offsets = (unsigned*)d_ws;               // NWIN+1
    unsigned* entries = offsets + (NWIN + 1);          // ntok entries

    build_buckets_kernel<<<1, 512, 0, stream>>>(idx, ntok, offsets, entries);

    dim3 grid(NWIN, VOCAB / VT);
    gather_scatter_kernel<<<grid, 256, 0, stream>>>(W, offsets, entries, out);
}